// LightGCN_17377437679754
// MI455X (gfx1250) — compile-verified
//
#include <hip/hip_runtime.h>

// ---------------- problem constants (from reference) ----------------
#define N_USERS 100000
#define N_NODES 150000
#define DIM     64
#define E_ELEMS (N_NODES * DIM)   // 9,600,000 floats = 38.4 MB per buffer

typedef float v2f __attribute__((ext_vector_type(2)));
typedef float v8f __attribute__((ext_vector_type(8)));

// ---------------- elementwise helpers (float4 vectorized) ----------------
__global__ void zero_f4(float4* __restrict__ p, int n4) {
    int i = blockIdx.x * blockDim.x + threadIdx.x;
    if (i < n4) p[i] = make_float4(0.f, 0.f, 0.f, 0.f);
}

__global__ void copy_f4(const float4* __restrict__ s, float4* __restrict__ d, int n4) {
    int i = blockIdx.x * blockDim.x + threadIdx.x;
    if (i < n4) d[i] = s[i];
}

__global__ void add_f4(const float4* __restrict__ s, float4* __restrict__ d, int n4) {
    int i = blockIdx.x * blockDim.x + threadIdx.x;
    if (i < n4) {
        float4 a = d[i], b = s[i];
        a.x += b.x; a.y += b.y; a.z += b.z; a.w += b.w;
        d[i] = a;
    }
}

// ---------------- COO SpMM: Eout[rows[e],:] += vals[e] * Ein[cols[e],:] ----------------
// 16 lanes per nonzero, float4 per lane (coalesced 256B row gather from L2,
// 4x global_atomic_add_f32 scatter handled by L2 atomic units).
__global__ void spmm_atomic(const int* __restrict__ rows,
                            const int* __restrict__ cols,
                            const float* __restrict__ vals,
                            const float* __restrict__ Ein,
                            float* __restrict__ Eout,
                            int nnz) {
    unsigned tid = blockIdx.x * blockDim.x + threadIdx.x;
    unsigned e = tid >> 4;             // nonzero index
    unsigned d = (tid & 15u) << 2;     // dim base (0,4,...,60)
    if (e >= (unsigned)nnz) return;
    int r = rows[e];
    int c = cols[e];
    float v = vals[e];
    const float4 x = *(const float4*)(Ein + (size_t)c * DIM + d);
    float* outp = Eout + (size_t)r * DIM + d;
    atomicAdd(outp + 0, v * x.x);
    atomicAdd(outp + 1, v * x.y);
    atomicAdd(outp + 2, v * x.z);
    atomicAdd(outp + 3, v * x.w);
}

// ---------------- batched dot via V_WMMA_F32_16X16X4_F32 ----------------
// One wave computes 16 (user,item) pair scores: C = A(16x64) x B(64x16) chained
// over 16 WMMA steps of K=4, then the diagonal of the 16x16 result is extracted
// per the CDNA5 C/D VGPR layout. Scale by 1/16 fuses the reference's /4 on
// both operands.
__global__ void dot_wmma(const float* __restrict__ acc,
                         const int* __restrict__ U,
                         const int* __restrict__ I,
                         float* __restrict__ out,
                         int ntiles) {
    int wave = (int)((blockIdx.x * blockDim.x + threadIdx.x) >> 5);
    int lane = (int)(threadIdx.x & 31u);
    if (wave >= ntiles) return;               // wave-uniform: EXEC stays all-1s

    int m  = lane & 15;          // A row / B col handled by this lane
    int kh = (lane >> 4) << 1;   // K sub-offset: lanes 0-15 -> 0, lanes 16-31 -> 2

    int b = wave * 16 + m;
    const float* rowU = acc + (size_t)U[b] * DIM;
    const float* rowI = acc + ((size_t)I[b] + N_USERS) * DIM;

    v8f c = {};
#pragma unroll
    for (int s = 0; s < 16; ++s) {
        int k = s * 4 + kh;
        v2f a  = *(const v2f*)(rowU + k);   // A[m][k..k+1]
        v2f bb = *(const v2f*)(rowI + k);   // B[k..k+1][m]
        // 8 args: (neg_a, A, neg_b, B, c_mod, C, reuse_a, reuse_b)
        c = __builtin_amdgcn_wmma_f32_16x16x4_f32(false, a, false, bb,
                                                  (short)0, c, false, false);
    }

    // Diagonal of 16x16 f32 C/D: lanes 0-7 hold C[l][l] in c[l];
    // lanes 24-31 hold C[l-16][l-16] in c[l-24].
    int j = lane & 7;
    int mrow = (lane < 16) ? lane : (lane - 16);
    if (lane < 8 || lane >= 24) {
        out[wave * 16 + mrow] = c[j] * 0.0625f;   // (1/4)*(1/4) = 1/16
    }
}

// ---------------- driver ----------------
extern "C" void kernel_launch(void* const* d_in, const int* in_sizes, int n_in,
                              void* d_out, int out_size, void* d_ws, size_t ws_size,
                              hipStream_t stream) {
    const float* emb    = (const float*)d_in[0];
    const int*   A_rows = (const int*)  d_in[1];
    const int*   A_cols = (const int*)  d_in[2];
    const float* A_vals = (const float*)d_in[3];
    const int*   U      = (const int*)  d_in[4];
    const int*   I      = (const int*)  d_in[5];
    float*       out    = (float*)d_out;

    const int nnz   = in_sizes[1];
    const int batch = in_sizes[4];

    float* Ea  = (float*)d_ws;
    float* Eb  = Ea + E_ELEMS;
    float* acc = Eb + E_ELEMS;

    const int T  = 256;
    const int n4 = E_ELEMS / 4;
    const int gElem = (n4 + T - 1) / T;
    const unsigned gSpmm = ((unsigned)nnz * 16u + T - 1) / T;

    // acc = E0
    copy_f4<<<gElem, T, 0, stream>>>((const float4*)emb, (float4*)acc, n4);

    // layer 1: Ea = A @ emb ; acc += Ea
    zero_f4<<<gElem, T, 0, stream>>>((float4*)Ea, n4);
    spmm_atomic<<<gSpmm, T, 0, stream>>>(A_rows, A_cols, A_vals, emb, Ea, nnz);
    add_f4<<<gElem, T, 0, stream>>>((const float4*)Ea, (float4*)acc, n4);

    // layer 2: Eb = A @ Ea ; acc += Eb
    zero_f4<<<gElem, T, 0, stream>>>((float4*)Eb, n4);
    spmm_atomic<<<gSpmm, T, 0, stream>>>(A_rows, A_cols, A_vals, Ea, Eb, nnz);
    add_f4<<<gElem, T, 0, stream>>>((const float4*)Eb, (float4*)acc, n4);

    // layer 3: Ea = A @ Eb ; acc += Ea
    zero_f4<<<gElem, T, 0, stream>>>((float4*)Ea, n4);
    spmm_atomic<<<gSpmm, T, 0, stream>>>(A_rows, A_cols, A_vals, Eb, Ea, nnz);
    add_f4<<<gElem, T, 0, stream>>>((const float4*)Ea, (float4*)acc, n4);

    // scoring: 16 pairs per wave, 8 waves per block
    const int ntiles = batch / 16;                 // 1024
    const int gDot   = (ntiles * 32 + T - 1) / T;  // 128 blocks
    dot_wmma<<<gDot, T, 0, stream>>>(acc, U, I, out, ntiles);
}